// ChebyshevConvolution_69879117905989
// MI455X (gfx1250) — compile-verified
//
#include <hip/hip_runtime.h>

#define N_NODES 100000

typedef __attribute__((ext_vector_type(2))) float v2f;
typedef __attribute__((ext_vector_type(8))) float v8f;

// ---------------------------------------------------------------------------
// Utility kernels
// ---------------------------------------------------------------------------
__global__ void zero_i32_kernel(int* __restrict__ p, int n) {
    int i = blockIdx.x * blockDim.x + threadIdx.x;
    if (i < n) p[i] = 0;
}

__global__ void copy_i32_kernel(const int* __restrict__ src,
                                int* __restrict__ dst, int n) {
    int i = blockIdx.x * blockDim.x + threadIdx.x;
    if (i < n) dst[i] = src[i];
}

// deg[dst] += 1 over all edges (integer)
__global__ void degree_kernel(const int* __restrict__ dst_idx,
                              int* __restrict__ deg, int E) {
    int e = blockIdx.x * blockDim.x + threadIdx.x;
    if (e < E) atomicAdd(&deg[dst_idx[e]], 1);
}

// dis = deg > 0 ? rsqrt(deg) : 0
__global__ void dis_kernel(const int* __restrict__ deg,
                           float* __restrict__ dis, int n) {
    int i = blockIdx.x * blockDim.x + threadIdx.x;
    if (i < n) {
        int d = deg[i];
        dis[i] = (d > 0) ? rsqrtf((float)d) : 0.0f;
    }
}

// ---------------------------------------------------------------------------
// CSR construction: degree -> exclusive prefix sum (rowptr) -> edge scatter.
// ---------------------------------------------------------------------------
__global__ void __launch_bounds__(256)
blocksum_kernel(const int* __restrict__ deg, int* __restrict__ bsum, int n) {
    __shared__ int sm[256];
    int i = blockIdx.x * 256 + threadIdx.x;
    sm[threadIdx.x] = (i < n) ? deg[i] : 0;
    __syncthreads();
#pragma unroll
    for (int s = 128; s > 0; s >>= 1) {
        if ((int)threadIdx.x < s) sm[threadIdx.x] += sm[threadIdx.x + s];
        __syncthreads();
    }
    if (threadIdx.x == 0) bsum[blockIdx.x] = sm[0];
}

// Single-block exclusive scan of block sums (nb <= 512)
__global__ void __launch_bounds__(512)
scan_bsum_kernel(int* __restrict__ bsum, int nb) {
    __shared__ int sm[512];
    int t = threadIdx.x;
    sm[t] = (t < nb) ? bsum[t] : 0;
    __syncthreads();
#pragma unroll
    for (int off = 1; off < 512; off <<= 1) {
        int v = (t >= off) ? sm[t - off] : 0;
        __syncthreads();
        sm[t] += v;
        __syncthreads();
    }
    int ex = (t == 0) ? 0 : sm[t - 1];
    if (t < nb) bsum[t] = ex;
}

// rowptr[i] = bsum[block] + exclusive-scan-in-block(deg); rowptr[n] = E
__global__ void __launch_bounds__(256)
rowptr_kernel(const int* __restrict__ deg, const int* __restrict__ bsum,
              int* __restrict__ rowptr, int n) {
    __shared__ int sm[256];
    int i = blockIdx.x * 256 + threadIdx.x;
    int d = (i < n) ? deg[i] : 0;
    sm[threadIdx.x] = d;
    __syncthreads();
#pragma unroll
    for (int off = 1; off < 256; off <<= 1) {
        int v = ((int)threadIdx.x >= off) ? sm[threadIdx.x - off] : 0;
        __syncthreads();
        sm[threadIdx.x] += v;
        __syncthreads();
    }
    int incl = sm[threadIdx.x];
    int base = bsum[blockIdx.x];
    if (i < n) rowptr[i] = base + incl - d;
    if (i == n - 1) rowptr[n] = base + incl;
}

// Scatter edges into CSR slots; precompute edge weight -dis[src]*dis[dst]
__global__ void __launch_bounds__(256)
scatter_kernel(const int* __restrict__ src_idx, const int* __restrict__ dst_idx,
               const float* __restrict__ dis, int* __restrict__ cursor,
               int* __restrict__ csr_src, float* __restrict__ csr_w, int E) {
    int e = blockIdx.x * blockDim.x + threadIdx.x;
    if (e >= E) return;
    int s = src_idx[e];
    int d = dst_idx[e];
    int slot = atomicAdd(&cursor[d], 1);
    csr_src[slot] = s;
    csr_w[slot] = -dis[s] * dis[d];
}

// ---------------------------------------------------------------------------
// CSR gather SpMM:  out[n,f] = scale * sum_{p in row n} w[p]*v[src[p], f]
//                              (+ beta * base[n,f] if ACCUM_BASE)
// One wave per (node, 32-feature chunk); register accumulation, single store,
// zero atomics. Gathered rows are L2-resident (<= 66MB << 192MB L2).
// Fuses T2 = 2*L_hat(T1) - T0 via (scale=2, beta=-1).
// ---------------------------------------------------------------------------
template <bool ACCUM_BASE>
__global__ void __launch_bounds__(256)
spmm_csr_kernel(const float* __restrict__ v, const float* __restrict__ base,
                float* __restrict__ out, const int* __restrict__ rowptr,
                const int* __restrict__ csr_src, const float* __restrict__ csr_w,
                int F, int fchunks, float scale, float beta, int nwaves) {
    int wid = (blockIdx.x * blockDim.x + threadIdx.x) >> 5;
    if (wid >= nwaves) return;
    int lane = threadIdx.x & 31;
    int n = wid / fchunks;
    int fc = wid - n * fchunks;
    int f = fc * 32 + lane;
    const bool act = (f < F);
    int p0 = rowptr[n];
    int p1 = rowptr[n + 1];
    float acc = 0.0f;
    for (int p = p0; p < p1; ++p) {
        int s = csr_src[p];
        float w = csr_w[p];
        if (act) acc += w * v[s * F + f];
    }
    if (act) {
        float r = scale * acc;
        if (ACCUM_BASE) r += beta * base[n * F + f];
        out[n * F + f] = r;
    }
}

// ---------------------------------------------------------------------------
// Repack W (3,K,N) row-major into WMMA B-fragment order, zero-padded:
//   Bp[((t*S + s)*2 + h)*NP + c] = { W[t][4s+2h][c], W[t][4s+2h+1][c] }
// ---------------------------------------------------------------------------
__global__ void pack_w_kernel(const float* __restrict__ W, float* __restrict__ Bp,
                              int K, int N, int S, int NP) {
    int i = blockIdx.x * blockDim.x + threadIdx.x;
    int total = 3 * S * 2 * NP;
    if (i >= total) return;
    int c = i % NP;
    int r = i / NP;          // r = (t*S + s)*2 + h
    int h = r & 1;
    int ts = r >> 1;
    int s = ts % S;
    int t = ts / S;
    int k = 4 * s + 2 * h;
    float x0 = (k < K && c < N) ? W[(t * K + k) * N + c] : 0.0f;
    float x1 = (k + 1 < K && c < N) ? W[(t * K + k + 1) * N + c] : 0.0f;
    Bp[2 * i]     = x0;
    Bp[2 * i + 1] = x1;
}

// ---------------------------------------------------------------------------
// Chebyshev GEMM:  out = relu6?(T0@W[0] + T1@W[1] + T2@W[2] + bias)
// One wave computes a 16-row node strip x NT*16 columns with
// V_WMMA_F32_16X16X4_F32. B pre-packed (unconditional coalesced v2f loads);
// only the single K%4 tail step (K=165) has guarded A loads.
// ---------------------------------------------------------------------------
template <int K, int NT, int N, bool RELU>
__global__ void __launch_bounds__(256)
gemm_cheb_kernel(const float* __restrict__ T0, const float* __restrict__ T1,
                 const float* __restrict__ T2, const float* __restrict__ Bp,
                 const float* __restrict__ bias, float* __restrict__ out,
                 int nstrips) {
    constexpr int S = (K + 3) / 4;   // K-steps of 4
    constexpr int NP = NT * 16;      // padded column count
    const int wave = threadIdx.x >> 5;
    const int lane = threadIdx.x & 31;
    const int strip = blockIdx.x * (blockDim.x >> 5) + wave;
    if (strip >= nstrips) return;    // wave-uniform: EXEC all-1s for WMMA

    const int half = lane >> 4;
    const int lrow = lane & 15;
    const int m0 = strip * 16;

    v8f acc[NT];
#pragma unroll
    for (int nt = 0; nt < NT; ++nt) acc[nt] = v8f{};

    const float* Ts[3] = {T0, T1, T2};
    for (int t = 0; t < 3; ++t) {
        const float* __restrict__ A = Ts[t];
        const int abase = (m0 + lrow) * K;          // 32-bit offsets -> saddr form
        const int bterm = t * S * 2 * NP * 2;       // float offset of term t
        for (int s = 0; s < S; ++s) {
            const int ka = 4 * s + 2 * half;
            v2f a;
            if constexpr ((K & 3) == 0) {
                a = *(const v2f*)(A + abase + ka);   // rows 8B-aligned (K even)
            } else {
                if (s == S - 1) {                    // only tail step guarded
                    a.x = (ka < K) ? A[abase + ka] : 0.0f;
                    a.y = (ka + 1 < K) ? A[abase + ka + 1] : 0.0f;
                } else {
                    a.x = A[abase + ka];
                    a.y = A[abase + ka + 1];
                }
            }
            const int boff = bterm + ((s * 2 + half) * NP + lrow) * 2;
#pragma unroll
            for (int nt = 0; nt < NT; ++nt) {
                v2f b = *(const v2f*)(Bp + boff + nt * 32);
                acc[nt] = __builtin_amdgcn_wmma_f32_16x16x4_f32(
                    false, a, false, b, (short)0, acc[nt], false, false);
            }
        }
    }

#pragma unroll
    for (int nt = 0; nt < NT; ++nt) {
        const int col = nt * 16 + lrow;
        if (N == NP || col < N) {
            const float bv = bias[col];
#pragma unroll
            for (int r = 0; r < 8; ++r) {
                float v = acc[nt][r] + bv;
                if (RELU) v = fminf(fmaxf(v, 0.0f), 6.0f);
                out[(m0 + r + 8 * half) * N + col] = v;
            }
        }
    }
}

// ---------------------------------------------------------------------------
// Host-side orchestration
// ---------------------------------------------------------------------------
static inline int cdiv(int a, int b) { return (a + b - 1) / b; }

extern "C" void kernel_launch(void* const* d_in, const int* in_sizes, int n_in,
                              void* d_out, int out_size, void* d_ws, size_t ws_size,
                              hipStream_t stream) {
    const float* x   = (const float*)d_in[0];
    const int*   ei  = (const int*)d_in[1];
    const int    E   = in_sizes[1] / 2;
    const int*   src = ei;
    const int*   dst = ei + E;
    const float* W1 = (const float*)d_in[2]; const float* b1 = (const float*)d_in[3];
    const float* W2 = (const float*)d_in[4]; const float* b2 = (const float*)d_in[5];
    const float* W3 = (const float*)d_in[6]; const float* b3 = (const float*)d_in[7];
    const float* W4 = (const float*)d_in[8]; const float* b4 = (const float*)d_in[9];

    const int NN = N_NODES;
    const int Fin = 165, H = 128;

    // Workspace layout (floats/ints; all offsets even -> 8B alignment kept)
    float* ws      = (float*)d_ws;
    float* dis     = ws;                        // NN
    float* csr_w   = dis + NN;                  // E
    float* T1      = csr_w + ((E + 1) & ~1);    // NN * Fin (max width)
    float* T2      = T1 + (size_t)NN * Fin;     // NN * Fin
    float* hA      = T2 + (size_t)NN * Fin;     // NN * H
    float* hB      = hA + (size_t)NN * H;       // NN * H
    float* Wpack   = hB + (size_t)NN * H;       // <= 3*42*2*128*2 floats
    int*   ideg    = (int*)(Wpack + 3 * 42 * 2 * 128 * 2); // NN
    int*   rowptr  = ideg + NN;                 // NN + 1
    int*   cursor  = rowptr + NN + 2;           // NN
    int*   csr_src = cursor + NN;               // E
    int*   bsum    = csr_src + E;               // cdiv(NN,256)

    const int TB = 256;
    const int nb = cdiv(NN, 256);               // 391 blocks of block-sums
    const int nstrips = NN / 16;                // 6250 (exact)
    const int gblk = cdiv(nstrips, TB / 32);

    // ---- CSR build (once; reused by all 8 propagations) ----
    zero_i32_kernel<<<cdiv(NN, TB), TB, 0, stream>>>(ideg, NN);
    degree_kernel<<<cdiv(E, TB), TB, 0, stream>>>(dst, ideg, E);
    dis_kernel<<<cdiv(NN, TB), TB, 0, stream>>>(ideg, dis, NN);
    blocksum_kernel<<<nb, 256, 0, stream>>>(ideg, bsum, NN);
    scan_bsum_kernel<<<1, 512, 0, stream>>>(bsum, nb);
    rowptr_kernel<<<nb, 256, 0, stream>>>(ideg, bsum, rowptr, NN);
    copy_i32_kernel<<<cdiv(NN, TB), TB, 0, stream>>>(rowptr, cursor, NN);
    scatter_kernel<<<cdiv(E, TB), TB, 0, stream>>>(src, dst, dis, cursor,
                                                   csr_src, csr_w, E);

    // T1 = L_hat(in); T2 = 2*L_hat(T1) - in   (gather-reduce, no atomics)
    auto cheb_terms = [&](const float* in, int F) {
        const int fchunks = (F + 31) / 32;
        const int nwaves = NN * fchunks;
        const int blocks = cdiv(nwaves, TB / 32);
        spmm_csr_kernel<false><<<blocks, TB, 0, stream>>>(
            in, in, T1, rowptr, csr_src, csr_w, F, fchunks, 1.0f, 0.0f, nwaves);
        spmm_csr_kernel<true><<<blocks, TB, 0, stream>>>(
            T1, in, T2, rowptr, csr_src, csr_w, F, fchunks, 2.0f, -1.0f, nwaves);
    };

    auto pack = [&](const float* W, int K, int N, int NP) {
        const int S = (K + 3) / 4;
        const int total = 3 * S * 2 * NP;
        pack_w_kernel<<<cdiv(total, TB), TB, 0, stream>>>(W, Wpack, K, N, S, NP);
    };

    // Layer 1: 165 -> 128, relu6
    cheb_terms(x, Fin);
    pack(W1, Fin, H, 128);
    gemm_cheb_kernel<165, 8, 128, true><<<gblk, TB, 0, stream>>>(
        x, T1, T2, Wpack, b1, hA, nstrips);

    // Layer 2: 128 -> 128, relu6
    cheb_terms(hA, H);
    pack(W2, H, H, 128);
    gemm_cheb_kernel<128, 8, 128, true><<<gblk, TB, 0, stream>>>(
        hA, T1, T2, Wpack, b2, hB, nstrips);

    // Layer 3: 128 -> 128, relu6
    cheb_terms(hB, H);
    pack(W3, H, H, 128);
    gemm_cheb_kernel<128, 8, 128, true><<<gblk, TB, 0, stream>>>(
        hB, T1, T2, Wpack, b3, hA, nstrips);

    // Layer 4: 128 -> 2, no activation; write directly to d_out
    cheb_terms(hA, H);
    pack(W4, H, 2, 16);
    gemm_cheb_kernel<128, 1, 2, false><<<gblk, TB, 0, stream>>>(
        hA, T1, T2, Wpack, b4, (float*)d_out, nstrips);

    // Tuple output: append edge_index after the (NN x 2) logits
    copy_i32_kernel<<<cdiv(in_sizes[1], TB), TB, 0, stream>>>(
        ei, (int*)d_out + (size_t)NN * 2, in_sizes[1]);
}